// RTDL_27857157881969
// MI455X (gfx1250) — compile-verified
//
#include <hip/hip_runtime.h>

#define BB   4096
#define NNUM 64
#define NCAT 64
#define CARD 100
#define DD   512
#define FF   128
#define KD   192   // dense-pooling GEMM K = 64 (x*w) + 64 (b_num) + 64 (b_cat)

typedef float v2f __attribute__((ext_vector_type(2)));
typedef float v8f __attribute__((ext_vector_type(8)));

// ---------------------------------------------------------------------------
// Kernel 1: dense pooling GEMM via fp32 WMMA.
// pooled_pre = A(16x192) x Wstacked(192x512) for BOTH masks (shared B frags).
// Coefficient matrices are prebuilt in LDS so the K-loop is branch-free.
// Grid: B/16 blocks x 256 threads (8 waves; each wave does 4 d-tiles of 16).
// ---------------------------------------------------------------------------
__global__ void k_pool_gemm(const float* __restrict__ x_num,
                            const float* __restrict__ mask_num,
                            const float* __restrict__ mask_cat,
                            const float* __restrict__ w_num,
                            const float* __restrict__ b_num,
                            const float* __restrict__ b_cat,
                            float* __restrict__ pooled_num,
                            float* __restrict__ pooled_cat) {
    __shared__ __align__(16) float sAn[16 * KD];
    __shared__ __align__(16) float sAc[16 * KD];
    const int b0  = blockIdx.x * 16;
    const int tid = threadIdx.x;

    // Build coefficient matrices: A[m][k] =
    //   k<64       : mask[m][k] * x_num[m][k]        (rows of w_num)
    //   64<=k<192  : mask[m][k-64]                   (rows of b_num / b_cat)
    for (int i = tid; i < 16 * KD; i += 256) {
        const int m = i / KD;
        const int k = i - m * KD;
        float vn, vc;
        if (k < 64) {
            const float x = x_num[(size_t)(b0 + m) * NNUM + k];
            vn = mask_num[(size_t)(b0 + m) * FF + k] * x;
            vc = mask_cat[(size_t)(b0 + m) * FF + k] * x;
        } else {
            vn = mask_num[(size_t)(b0 + m) * FF + (k - 64)];
            vc = mask_cat[(size_t)(b0 + m) * FF + (k - 64)];
        }
        sAn[i] = vn;
        sAc[i] = vc;
    }
    __syncthreads();

    const int lane = tid & 31;
    const int wave = tid >> 5;
    const int half = lane >> 4;
    const int n    = lane & 15;

    // per-lane A row (M = n); pairs (kk,kk+1) are contiguous -> ds_load_b64
    const v2f* An = (const v2f*)(sAn + n * KD);
    const v2f* Ac = (const v2f*)(sAc + n * KD);

    for (int it = 0; it < 4; ++it) {
        const int d0 = (wave + 8 * it) * 16;
        const float* wb[3] = { w_num + d0 + n, b_num + d0 + n, b_cat + d0 + n };
        v8f cn = {};
        v8f cc = {};
#pragma unroll
        for (int reg = 0; reg < 3; ++reg) {      // fully unrolled: no pointer branches
            const float* w = wb[reg];
            const int kb = reg * 64;
#pragma unroll 4
            for (int k0 = 0; k0 < 64; k0 += 4) {
                const int kl = k0 + half * 2;            // local k of this lane's pair
                const v2f an = An[(kb + kl) >> 1];
                const v2f ac = Ac[(kb + kl) >> 1];
                v2f bf;
                bf[0] = w[(size_t)kl * DD];
                bf[1] = w[(size_t)(kl + 1) * DD];
                cn = __builtin_amdgcn_wmma_f32_16x16x4_f32(false, an, false, bf, (short)0, cn, false, false);
                cc = __builtin_amdgcn_wmma_f32_16x16x4_f32(false, ac, false, bf, (short)0, cc, false, false);
            }
        }
#pragma unroll
        for (int r = 0; r < 8; ++r) {
            const int m = r + half * 8;
            pooled_num[(size_t)(b0 + m) * DD + d0 + n] = cn[r];
            pooled_cat[(size_t)(b0 + m) * DD + d0 + n] = cc[r];
        }
    }
}

// ---------------------------------------------------------------------------
// Kernel 2: per-row embedding gather (irregular -> VALU), add to pre-pooled,
// then normalize by mask sums. Grid: B blocks x 128 threads (float4/thread).
// ---------------------------------------------------------------------------
__global__ void k_gather_norm(const int* __restrict__ x_cat,
                              const float* __restrict__ mask_num,
                              const float* __restrict__ mask_cat,
                              const float* __restrict__ emb_cat,
                              float* __restrict__ pooled_num,
                              float* __restrict__ pooled_cat) {
    const int b = blockIdx.x;
    const int d = threadIdx.x * 4;
    float4 accn = *(const float4*)(pooled_num + (size_t)b * DD + d);
    float4 accc = *(const float4*)(pooled_cat + (size_t)b * DD + d);

    float smn = 0.f, smc = 0.f;
    for (int f = 0; f < FF; ++f) {   // uniform scalar loads (b is block-uniform)
        smn += mask_num[(size_t)b * FF + f];
        smc += mask_cat[(size_t)b * FF + f];
    }
    for (int c = 0; c < NCAT; ++c) {
        const float sn = mask_num[(size_t)b * FF + 64 + c];
        const float sc = mask_cat[(size_t)b * FF + 64 + c];
        if (sn != 0.f || sc != 0.f) {
            const int row = x_cat[(size_t)b * NCAT + c] + c * CARD;
            const float4 ev = *(const float4*)(emb_cat + (size_t)row * DD + d);
            accn.x += sn * ev.x; accn.y += sn * ev.y; accn.z += sn * ev.z; accn.w += sn * ev.w;
            accc.x += sc * ev.x; accc.y += sc * ev.y; accc.z += sc * ev.z; accc.w += sc * ev.w;
        }
    }
    const float in = 1.f / smn, ic = 1.f / smc;
    accn.x *= in; accn.y *= in; accn.z *= in; accn.w *= in;
    accc.x *= ic; accc.y *= ic; accc.z *= ic; accc.w *= ic;
    *(float4*)(pooled_num + (size_t)b * DD + d) = accn;
    *(float4*)(pooled_cat + (size_t)b * DD + d) = accc;
}

// ---------------------------------------------------------------------------
// Kernel 3: pred_num — per-sample dot(pooled_num[b], dec_num_W[idx]) + bias.
// One wave per sample, float4 (b128) loads; 4 samples per 128-thread block.
// ---------------------------------------------------------------------------
__global__ void k_pred_num(const float* __restrict__ pooled_num,
                           const int* __restrict__ pred_idx_num,
                           const float* __restrict__ dec_num_W,
                           const float* __restrict__ dec_num_b,
                           float* __restrict__ out) {
    const int lane = threadIdx.x & 31;
    const int b = blockIdx.x * 4 + (threadIdx.x >> 5);
    const int e = pred_idx_num[b];
    const float4* p = (const float4*)(pooled_num + (size_t)b * DD) + lane * 4;
    const float4* w = (const float4*)(dec_num_W  + (size_t)e * DD) + lane * 4;
    float acc = 0.f;
#pragma unroll
    for (int j = 0; j < 4; ++j) {
        const float4 a = p[j];
        const float4 c = w[j];
        acc += a.x * c.x + a.y * c.y + a.z * c.z + a.w * c.w;
    }
#pragma unroll
    for (int s = 16; s > 0; s >>= 1) acc += __shfl_xor(acc, s, 32);
    if (lane == 0) out[b] = acc + dec_num_b[e];
}

// ------------------------- expert grouping (tiny) --------------------------
__global__ void k_init(int* counts) { counts[threadIdx.x] = 0; }

__global__ void k_count(const int* __restrict__ pic, int* counts) {
    const int b = blockIdx.x * 256 + threadIdx.x;
    atomicAdd(&counts[pic[b] - NNUM], 1);
}

__global__ void k_scan_tiles(const int* __restrict__ counts, int* offsets, int* fill,
                             int* tileE, int* tileS, int* tileV, int* tileCount) {
    if (threadIdx.x != 0) return;
    int acc = 0;
    for (int e = 0; e < NCAT; ++e) { offsets[e] = acc; fill[e] = acc; acc += counts[e]; }
    offsets[NCAT] = acc;
    int tc = 0;
    for (int e = 0; e < NCAT; ++e) {
        const int s = offsets[e], cnt = counts[e];
        for (int t = 0; t < cnt; t += 16) {
            tileE[tc] = e;
            tileS[tc] = s + t;
            tileV[tc] = (cnt - t < 16) ? (cnt - t) : 16;
            ++tc;
        }
    }
    *tileCount = tc;
}

__global__ void k_scatter(const int* __restrict__ pic, int* fill, int* perm) {
    const int b = blockIdx.x * 256 + threadIdx.x;
    const int e = pic[b] - NNUM;
    perm[atomicAdd(&fill[e], 1)] = b;
}

// ---------------------------------------------------------------------------
// Kernel 5: pred_cat via fp32 WMMA on expert-grouped 16-sample tiles.
// A and B pairs are contiguous in k -> b64 loads; K=512 -> 128 v_wmma/block.
// Grid: (maxTiles, 7) blocks of 1 wave.
// ---------------------------------------------------------------------------
__global__ void k_pred_cat(const float* __restrict__ pooled_cat,
                           const float* __restrict__ dec_cat_W,
                           const float* __restrict__ dec_cat_b,
                           const int* __restrict__ perm,
                           const int* __restrict__ tileE,
                           const int* __restrict__ tileS,
                           const int* __restrict__ tileV,
                           const int* __restrict__ tileCount,
                           float* __restrict__ out_cat) {
    const int tile = blockIdx.x;
    if (tile >= *tileCount) return;
    const int e  = tileE[tile];
    const int s0 = tileS[tile];
    const int nv = tileV[tile];
    const int c0 = blockIdx.y * 16;

    const int lane = threadIdx.x & 31;
    const int half = lane >> 4;
    const int n    = lane & 15;

    const int brow = perm[s0 + (n < nv ? n : nv - 1)];
    const int ccol = (c0 + n < CARD) ? (c0 + n) : (CARD - 1);
    const v2f* pa = (const v2f*)(pooled_cat + (size_t)brow * DD) + half;          // [k0/2 + half]
    const v2f* pw = (const v2f*)(dec_cat_W + ((size_t)e * CARD + ccol) * DD) + half;

    v8f acc = {};
#pragma unroll 4
    for (int k0 = 0; k0 < DD; k0 += 4) {
        const v2f a  = pa[k0 >> 1];
        const v2f bf = pw[k0 >> 1];
        acc = __builtin_amdgcn_wmma_f32_16x16x4_f32(false, a, false, bf, (short)0, acc, false, false);
    }
#pragma unroll
    for (int r = 0; r < 8; ++r) {
        const int m = r + half * 8;
        if (m < nv && (c0 + n) < CARD) {
            const int b = perm[s0 + m];
            out_cat[(size_t)b * CARD + c0 + n] = acc[r] + dec_cat_b[e * CARD + c0 + n];
        }
    }
}

// ---------------------------------------------------------------------------
extern "C" void kernel_launch(void* const* d_in, const int* in_sizes, int n_in,
                              void* d_out, int out_size, void* d_ws, size_t ws_size,
                              hipStream_t stream) {
    (void)in_sizes; (void)n_in; (void)out_size; (void)ws_size;
    const float* x_num        = (const float*)d_in[0];
    const int*   x_cat        = (const int*)  d_in[1];
    const float* mask_num     = (const float*)d_in[2];
    const float* mask_cat     = (const float*)d_in[3];
    const int*   pred_idx_num = (const int*)  d_in[4];
    const int*   pred_idx_cat = (const int*)  d_in[5];
    const float* w_num        = (const float*)d_in[6];
    const float* b_num        = (const float*)d_in[7];
    const float* emb_cat      = (const float*)d_in[8];
    const float* b_cat        = (const float*)d_in[9];
    const float* dec_num_W    = (const float*)d_in[10];
    const float* dec_num_b    = (const float*)d_in[11];
    const float* dec_cat_W    = (const float*)d_in[12];
    const float* dec_cat_b    = (const float*)d_in[13];
    float* out = (float*)d_out;

    // workspace carve-up (~16.8 MB)
    float* pooled_num = (float*)d_ws;
    float* pooled_cat = pooled_num + (size_t)BB * DD;
    int* counts    = (int*)(pooled_cat + (size_t)BB * DD);
    int* offsets   = counts + 64;
    int* fill      = offsets + 65;
    int* perm      = fill + 64;
    int* tileE     = perm + BB;
    int* tileS     = tileE + 384;
    int* tileV     = tileS + 384;
    int* tileCount = tileV + 384;

    hipLaunchKernelGGL(k_init, dim3(1), dim3(64), 0, stream, counts);
    hipLaunchKernelGGL(k_count, dim3(BB / 256), dim3(256), 0, stream, pred_idx_cat, counts);
    hipLaunchKernelGGL(k_scan_tiles, dim3(1), dim3(32), 0, stream,
                       counts, offsets, fill, tileE, tileS, tileV, tileCount);
    hipLaunchKernelGGL(k_scatter, dim3(BB / 256), dim3(256), 0, stream, pred_idx_cat, fill, perm);

    hipLaunchKernelGGL(k_pool_gemm, dim3(BB / 16), dim3(256), 0, stream,
                       x_num, mask_num, mask_cat, w_num, b_num, b_cat,
                       pooled_num, pooled_cat);
    hipLaunchKernelGGL(k_gather_norm, dim3(BB), dim3(128), 0, stream,
                       x_cat, mask_num, mask_cat, emb_cat, pooled_num, pooled_cat);

    hipLaunchKernelGGL(k_pred_num, dim3(BB / 4), dim3(128), 0, stream,
                       pooled_num, pred_idx_num, dec_num_W, dec_num_b, out);
    hipLaunchKernelGGL(k_pred_cat, dim3(384, 7), dim3(32), 0, stream,
                       pooled_cat, dec_cat_W, dec_cat_b, perm,
                       tileE, tileS, tileV, tileCount, out + BB);
}